// GraphConvolution_13082470384220
// MI455X (gfx1250) — compile-verified
//
#include <hip/hip_runtime.h>
#include <hip/hip_bf16.h>

// GraphConvolution for MI455X (gfx1250, wave32).
//   support = X @ W          (WMMA f32 16x16x4; HBM-bound ~200MB)
//   out     = bias broadcast
//   out[r] += val * support[c]  per edge (L2-resident gather + f32 atomics)

#define GC_N_NODES 100000
#define GC_D       256        // D_IN == D_OUT
#define GC_LDS_PITCH (GC_D + 4)   // +4 floats: A-fragment reads hit 16 banks

typedef float v2f __attribute__((ext_vector_type(2)));
typedef float v8f __attribute__((ext_vector_type(8)));

// ---------------------------------------------------------------------------
// Kernel 1: support = X @ W  using V_WMMA_F32_16X16X4_F32.
// Block = 256 threads (8 waves), M-tile = 32 rows (halves W L2 re-reads vs
// M=16: 0.8GB total). Wave w owns cols [w*32, w*32+32) for BOTH 16-row
// subtiles -> 4 accumulators sharing one B fragment per k-step.
// ---------------------------------------------------------------------------
__global__ __launch_bounds__(256) void gc_gemm_wmma_f32(
    const float* __restrict__ X,      // [N, 256] row-major
    const float* __restrict__ W,      // [256, 256] row-major
    float* __restrict__ support) {    // [N, 256]
  __shared__ float Xs[32][GC_LDS_PITCH];

  const int tid   = threadIdx.x;
  const int lane  = tid & 31;
  const int wave  = tid >> 5;
  const int l16   = lane & 15;      // position within half-wave
  const int half  = lane >> 4;      // 0: K=0/1 side, 1: K=2/3 side (A layout)
  const int mBase = blockIdx.x * 32;
  const int nBase = wave * 32;

  // Warm L2/L0 with the W panel (emits global_prefetch_b8).
  __builtin_prefetch(W + (size_t)tid * (GC_D * GC_D / 256), 0, 2);

  // ---- Stage 32x256 X tile via gfx1250 async-to-LDS (ASYNCcnt-tracked) ----
  // thread t -> row t/8, 32 floats (8 x b128). 256 threads x 128B = 32KB.
  // IMPORTANT: the LDS-address operand is derived from &Xs (ptrtoint of the
  // generic pointer; low 32 bits == workgroup-relative LDS offset per the
  // aperture rules). This captures Xs so the "memory" clobber makes the
  // compiler treat the asm as a writer of Xs -- otherwise the A-fragment
  // ds_loads get folded to undef (observed in the previous round's disasm).
  {
    const int r  = tid >> 3;          // 0..31
    const int c0 = (tid & 7) * 32;    // 0..224
    const float* gsrc = X + (size_t)(mBase + r) * GC_D + c0;
    float* lds = &Xs[r][c0];
    #pragma unroll
    for (int j = 0; j < 8; ++j) {
      unsigned dst = (unsigned)(uintptr_t)(lds + j * 4);        // j*16 bytes
      unsigned long long src = (unsigned long long)(uintptr_t)(gsrc + j * 4);
      asm volatile("global_load_async_to_lds_b128 %0, %1, off"
                   :: "v"(dst), "v"(src) : "memory");
    }
    asm volatile("s_wait_asynccnt 0x0" ::: "memory");
  }
  __syncthreads();

  v8f accL0 = {};   // rows [mBase,    +16), cols [nBase,    +16)
  v8f accL1 = {};   // rows [mBase,    +16), cols [nBase+16, +32)
  v8f accH0 = {};   // rows [mBase+16, +32), cols [nBase,    +16)
  v8f accH1 = {};   // rows [mBase+16, +32), cols [nBase+16, +32)

  #pragma unroll 4
  for (int k = 0; k < GC_D; k += 4) {
    // A fragments 16x4 (ISA 7.12.2): lanes 0-15 rows 0-15;
    //   VGPR0 = {K=0 | K=2}, VGPR1 = {K=1 | K=3}.
    v2f aL, aH;
    aL.x = Xs[l16][k + half * 2 + 0];
    aL.y = Xs[l16][k + half * 2 + 1];
    aH.x = Xs[l16 + 16][k + half * 2 + 0];
    aH.y = Xs[l16 + 16][k + half * 2 + 1];

    // B fragment 4x16 (shared by both M-subtiles): VGPR0 = {K=0 | K=2},
    // VGPR1 = {K=1 | K=3}; row striped across lanes. 64B coalesced/half-wave.
    const float* w0 = W + (size_t)(k + half * 2 + 0) * GC_D + nBase;
    const float* w1 = W + (size_t)(k + half * 2 + 1) * GC_D + nBase;
    v2f b0, b1;
    b0.x = w0[l16];       b0.y = w1[l16];
    b1.x = w0[l16 + 16];  b1.y = w1[l16 + 16];

    accL0 = __builtin_amdgcn_wmma_f32_16x16x4_f32(
        false, aL, false, b0, (short)0, accL0, false, false);
    accL1 = __builtin_amdgcn_wmma_f32_16x16x4_f32(
        false, aL, false, b1, (short)0, accL1, false, false);
    accH0 = __builtin_amdgcn_wmma_f32_16x16x4_f32(
        false, aH, false, b0, (short)0, accH0, false, false);
    accH1 = __builtin_amdgcn_wmma_f32_16x16x4_f32(
        false, aH, false, b1, (short)0, accH1, false, false);
  }

  // C/D layout: VGPR i -> lanes 0-15: M=i, N=lane; lanes 16-31: M=i+8.
  #pragma unroll
  for (int i = 0; i < 8; ++i) {
    const int rowL = mBase + i + half * 8;
    float* dstL = support + (size_t)rowL * GC_D + nBase + l16;
    dstL[0]  = accL0[i];
    dstL[16] = accL1[i];
    float* dstH = dstL + 16 * GC_D;       // rows mBase+16 ..
    dstH[0]  = accH0[i];
    dstH[16] = accH1[i];
  }
}

// ---------------------------------------------------------------------------
// Kernel 2: out[n][d] = bias[d]  (bias folded into the accumulator init)
// ---------------------------------------------------------------------------
__global__ __launch_bounds__(256) void gc_init_bias(
    const float* __restrict__ bias, float* __restrict__ out, int total4) {
  int idx = blockIdx.x * 256 + threadIdx.x;              // float4 index
  if (idx < total4) {
    const float4* b4 = reinterpret_cast<const float4*>(bias);
    reinterpret_cast<float4*>(out)[idx] = b4[idx & (GC_D / 4 - 1)];
  }
}

// ---------------------------------------------------------------------------
// Kernel 3: COO scatter.  One wave per edge: lane l gathers 8 columns of
// support[col] (two float4 L2 hits) and issues 8 non-returning f32 atomics.
// ---------------------------------------------------------------------------
__global__ __launch_bounds__(256) void gc_spmm_scatter(
    const float* __restrict__ support,
    const float* __restrict__ evals,
    const int*   __restrict__ erows,
    const int*   __restrict__ ecols,
    float* __restrict__ out, int nEdges) {
  const int eid  = blockIdx.x * 8 + (threadIdx.x >> 5);
  const int lane = threadIdx.x & 31;
  if (eid >= nEdges) return;

  const int   r = erows[eid];
  const int   c = ecols[eid];
  const float v = evals[eid];

  const float4* src = reinterpret_cast<const float4*>(support + (size_t)c * GC_D);
  float4 s0 = src[lane];        // cols [lane*4, lane*4+4)
  float4 s1 = src[lane + 32];   // cols [128+lane*4, ...)

  float* dst0 = out + (size_t)r * GC_D + lane * 4;
  float* dst1 = dst0 + 128;
  __hip_atomic_fetch_add(dst0 + 0, v * s0.x, __ATOMIC_RELAXED, __HIP_MEMORY_SCOPE_AGENT);
  __hip_atomic_fetch_add(dst0 + 1, v * s0.y, __ATOMIC_RELAXED, __HIP_MEMORY_SCOPE_AGENT);
  __hip_atomic_fetch_add(dst0 + 2, v * s0.z, __ATOMIC_RELAXED, __HIP_MEMORY_SCOPE_AGENT);
  __hip_atomic_fetch_add(dst0 + 3, v * s0.w, __ATOMIC_RELAXED, __HIP_MEMORY_SCOPE_AGENT);
  __hip_atomic_fetch_add(dst1 + 0, v * s1.x, __ATOMIC_RELAXED, __HIP_MEMORY_SCOPE_AGENT);
  __hip_atomic_fetch_add(dst1 + 1, v * s1.y, __ATOMIC_RELAXED, __HIP_MEMORY_SCOPE_AGENT);
  __hip_atomic_fetch_add(dst1 + 2, v * s1.z, __ATOMIC_RELAXED, __HIP_MEMORY_SCOPE_AGENT);
  __hip_atomic_fetch_add(dst1 + 3, v * s1.w, __ATOMIC_RELAXED, __HIP_MEMORY_SCOPE_AGENT);
}

// ---------------------------------------------------------------------------
extern "C" void kernel_launch(void* const* d_in, const int* in_sizes, int n_in,
                              void* d_out, int out_size, void* d_ws, size_t ws_size,
                              hipStream_t stream) {
  const float* X     = (const float*)d_in[0];  // [100000,256]
  const float* W     = (const float*)d_in[1];  // [256,256]
  const float* bias  = (const float*)d_in[2];  // [256]
  const float* evals = (const float*)d_in[3];  // [E]
  const int*   erows = (const int*)  d_in[4];  // [E]
  const int*   ecols = (const int*)  d_in[5];  // [E]
  float* out     = (float*)d_out;
  float* support = (float*)d_ws;               // needs N*D*4 = 102.4 MB scratch
  const int nEdges = in_sizes[3];

  // 1) support = X @ W   (3125 M-tiles of 32 rows; 100000 % 32 == 0)
  gc_gemm_wmma_f32<<<GC_N_NODES / 32, 256, 0, stream>>>(X, W, support);

  // 2) out = broadcast(bias)
  const int total4 = GC_N_NODES * GC_D / 4;    // 6.4M float4
  gc_init_bias<<<(total4 + 255) / 256, 256, 0, stream>>>(bias, out, total4);

  // 3) out[r] += val * support[c]  (one wave per edge)
  gc_spmm_scatter<<<(nEdges + 7) / 8, 256, 0, stream>>>(
      support, evals, erows, ecols, out, nEdges);
}